// GCNClassifier_34557306864325
// MI455X (gfx1250) — compile-verified
//
#include <hip/hip_runtime.h>
#include <hip/hip_bf16.h>
#include <stdint.h>

#define N_NODES 50000
#define N_EDGES 800000
#define D 128   // D_IN == D_HID == 128

typedef __attribute__((ext_vector_type(2))) float v2f;
typedef __attribute__((ext_vector_type(8))) float v8f;

// ---------------------------------------------------------------------------
// deg[i] = 1.0 (self loop)
__global__ __launch_bounds__(256) void k_init_deg(float* __restrict__ deg) {
    int i = blockIdx.x * 256 + threadIdx.x;
    if (i < N_NODES) deg[i] = 1.0f;
}

// deg[row[e]] += 1
__global__ __launch_bounds__(256) void k_count_deg(const long long* __restrict__ ei,
                                                   float* __restrict__ deg) {
    int e = blockIdx.x * 256 + threadIdx.x;
    if (e < N_EDGES) atomicAdd(&deg[(int)ei[e]], 1.0f);
}

// dinv[i] = rsqrt(deg[i])   (deg >= 1 always, so no zero guard needed)
__global__ __launch_bounds__(256) void k_dinv(float* __restrict__ degdinv) {
    int i = blockIdx.x * 256 + threadIdx.x;
    if (i < N_NODES) degdinv[i] = rsqrtf(degdinv[i]);
}

// ---------------------------------------------------------------------------
// support = x @ W1^T + b1 using V_WMMA_F32_16X16X4_F32 (exact fp32 path).
// Block = 256 threads = 8 waves. Block -> 16 node rows; wave w -> cols [16w,16w+16).
// A (16x4 f32): lane L holds M=L&15, K={kh, kh+1}, kh = (L>=16 ? 2 : 0).
// B (4x16)  : lane L holds N=L&15, same K pair.  B[k][n] = W1[n][k].
// C/D (8 VGPR): acc[r] is M = r + (L>=16 ? 8 : 0), N = L&15.
__global__ __launch_bounds__(256) void k_gemm1_wmma(const float* __restrict__ x,
                                                    const float* __restrict__ W1,
                                                    const float* __restrict__ b1,
                                                    float* __restrict__ support) {
    const int lane  = threadIdx.x & 31;
    const int wave  = threadIdx.x >> 5;
    const int row0  = blockIdx.x * 16;
    const int col0  = wave * 16;
    const int mn    = lane & 15;
    const int khalf = (lane >> 4) << 1;   // 0 or 2

    const float bias = b1[col0 + mn];
    v8f acc;
#pragma unroll
    for (int r = 0; r < 8; ++r) acc[r] = bias;

    const float* arow = x  + (size_t)(row0 + mn) * D;   // node row
    const float* brow = W1 + (size_t)(col0 + mn) * D;   // W1 row (gives B^T access)

#pragma unroll 4
    for (int kb = 0; kb < D; kb += 4) {
        v2f a = *(const v2f*)(arow + kb + khalf);
        v2f b = *(const v2f*)(brow + kb + khalf);
        // 8 args: (neg_a, A, neg_b, B, c_mod, C, reuse_a, reuse_b)
        acc = __builtin_amdgcn_wmma_f32_16x16x4_f32(false, a, false, b,
                                                    (short)0, acc, false, false);
    }

    const int mbase = row0 + ((lane >> 4) << 3);
#pragma unroll
    for (int r = 0; r < 8; ++r)
        support[(size_t)(mbase + r) * D + col0 + mn] = acc[r];
}

// ---------------------------------------------------------------------------
// h[i][:] = dinv[i]^2 * support[i][:]   (self-loop contribution, also zero-inits h)
__global__ __launch_bounds__(256) void k_init_h(const float* __restrict__ dinv,
                                                const float* __restrict__ support,
                                                float* __restrict__ h) {
    int idx = blockIdx.x * 256 + threadIdx.x;          // float4 index
    const int nvec = N_NODES * (D / 4);
    if (idx >= nvec) return;
    int node = idx >> 5;                               // 32 float4 per node
    float s = dinv[node]; s *= s;
    float4 v = ((const float4*)support)[idx];
    v.x *= s; v.y *= s; v.z *= s; v.w *= s;
    ((float4*)h)[idx] = v;
}

// h[row[e]][:] += dinv[r]*dinv[c] * support[col[e]][:]   — one wave per edge
__global__ __launch_bounds__(256) void k_scatter1(const long long* __restrict__ ei,
                                                  const float* __restrict__ dinv,
                                                  const float* __restrict__ support,
                                                  float* __restrict__ h) {
    int e = blockIdx.x * 8 + (threadIdx.x >> 5);
    if (e >= N_EDGES) return;
    int lane = threadIdx.x & 31;
    int r = (int)ei[e];
    int c = (int)ei[N_EDGES + e];
    float nrm = dinv[r] * dinv[c];
    float4 v = ((const float4*)(support + (size_t)c * D))[lane];
    float* dst = h + (size_t)r * D + lane * 4;
    atomicAdd(dst + 0, nrm * v.x);
    atomicAdd(dst + 1, nrm * v.y);
    atomicAdd(dst + 2, nrm * v.z);
    atomicAdd(dst + 3, nrm * v.w);
}

// support2[i] = relu(h[i]) . W2 + b2   — one wave per node, wave32 shuffle reduce
__global__ __launch_bounds__(256) void k_layer2_dot(const float* __restrict__ h,
                                                    const float* __restrict__ W2,
                                                    const float* __restrict__ b2,
                                                    float* __restrict__ support2) {
    int i = blockIdx.x * 8 + (threadIdx.x >> 5);
    if (i >= N_NODES) return;
    int lane = threadIdx.x & 31;
    float4 hv = ((const float4*)(h + (size_t)i * D))[lane];
    float4 wv = ((const float4*)W2)[lane];
    float s = fmaxf(hv.x, 0.f) * wv.x + fmaxf(hv.y, 0.f) * wv.y +
              fmaxf(hv.z, 0.f) * wv.z + fmaxf(hv.w, 0.f) * wv.w;
#pragma unroll
    for (int off = 16; off > 0; off >>= 1)
        s += __shfl_xor(s, off, 32);
    if (lane == 0) support2[i] = s + b2[0];
}

// out[i] = dinv[i]^2 * support2[i]   (self loop; overwrites poisoned d_out)
__global__ __launch_bounds__(256) void k_init_logits(const float* __restrict__ dinv,
                                                     const float* __restrict__ support2,
                                                     float* __restrict__ out) {
    int i = blockIdx.x * 256 + threadIdx.x;
    if (i < N_NODES) {
        float s = dinv[i]; s *= s;
        out[i] = s * support2[i];
    }
}

// out[row[e]] += dinv[r]*dinv[c] * support2[col[e]]
__global__ __launch_bounds__(256) void k_scatter2(const long long* __restrict__ ei,
                                                  const float* __restrict__ dinv,
                                                  const float* __restrict__ support2,
                                                  float* __restrict__ out) {
    int e = blockIdx.x * 256 + threadIdx.x;
    if (e >= N_EDGES) return;
    int r = (int)ei[e];
    int c = (int)ei[N_EDGES + e];
    atomicAdd(&out[r], dinv[r] * dinv[c] * support2[c]);
}

// ---------------------------------------------------------------------------
extern "C" void kernel_launch(void* const* d_in, const int* in_sizes, int n_in,
                              void* d_out, int out_size, void* d_ws, size_t ws_size,
                              hipStream_t stream) {
    const float*      x  = (const float*)d_in[0];
    const long long*  ei = (const long long*)d_in[1];   // int64 [2, E]
    const float*      W1 = (const float*)d_in[2];       // [128,128]
    const float*      b1 = (const float*)d_in[3];       // [128]
    const float*      W2 = (const float*)d_in[4];       // [1,128]
    const float*      b2 = (const float*)d_in[5];       // [1]
    float* out = (float*)d_out;                         // [50000]

    // workspace layout (floats); 50016 keeps 16B alignment for float4 views
    float* dinv     = (float*)d_ws;                      // 50000  (deg, then dinv)
    float* support  = dinv + 50016;                      // 50000*128
    float* h        = support + (size_t)N_NODES * D;     // 50000*128
    float* support2 = h + (size_t)N_NODES * D;           // 50000

    const int nblk_nodes = (N_NODES + 255) / 256;        // 196
    const int nblk_edges = (N_EDGES + 255) / 256;        // 3125

    k_init_deg   <<<nblk_nodes, 256, 0, stream>>>(dinv);
    k_count_deg  <<<nblk_edges, 256, 0, stream>>>(ei, dinv);
    k_dinv       <<<nblk_nodes, 256, 0, stream>>>(dinv);

    k_gemm1_wmma <<<N_NODES / 16, 256, 0, stream>>>(x, W1, b1, support);  // 3125 blocks

    k_init_h     <<<(N_NODES * (D / 4) + 255) / 256, 256, 0, stream>>>(dinv, support, h);
    k_scatter1   <<<(N_EDGES + 7) / 8, 256, 0, stream>>>(ei, dinv, support, h);

    k_layer2_dot <<<(N_NODES + 7) / 8, 256, 0, stream>>>(h, W2, b2, support2);

    k_init_logits<<<nblk_nodes, 256, 0, stream>>>(dinv, support2, out);
    k_scatter2   <<<nblk_edges, 256, 0, stream>>>(ei, dinv, support2, out);
}